// LeafLogMixtureHead_16630113370385
// MI455X (gfx1250) — compile-verified
//
#include <hip/hip_runtime.h>
#include <math.h>

// y = log( clip(mu,1e-12) @ softmax(leaf_scores, axis=1) )
// mu: (32768, 256) f32   leaf_scores: (256, 64) f32   out: (32768, 64) f32
// HBM-bound (~40MB @ 23.3TB/s); fp32 WMMA 16x16x4 does the math for free.

typedef float v2f __attribute__((ext_vector_type(2)));
typedef float v8f __attribute__((ext_vector_type(8)));

#define N_ROWS   32768
#define L_DIM    256        // K (leaves)
#define C_DIM    64         // output classes
#define LDSTRIDE 260        // padded row stride (260 % 64 == 4) -> conflict-free ds_load_b64
#define ROWS_PER_BLOCK 128  // 8 waves x 16 rows

__global__ __launch_bounds__(256) void LeafLogMixtureHead_kernel(
        const float* __restrict__ mu,     // N x 256 row-major
        const float* __restrict__ leaf,   // 256 x 64 row-major
        float* __restrict__ out) {        // N x 64 row-major
    extern __shared__ float ldsQ[];       // Qt: 64 rows x LDSTRIDE floats (transposed softmax)

    // ---- Stage Qt = softmax(leaf)^T into LDS: one leaf-row per thread ----
    {
        const int l = threadIdx.x;                 // 0..255
        const float* srow = leaf + l * C_DIM;
        float m = -3.402823466e+38f;
        #pragma unroll
        for (int c = 0; c < C_DIM; ++c) m = fmaxf(m, srow[c]);
        float sum = 0.0f;
        #pragma unroll
        for (int c = 0; c < C_DIM; ++c) sum += expf(srow[c] - m);
        const float inv = 1.0f / sum;
        #pragma unroll
        for (int c = 0; c < C_DIM; ++c)            // column l of each padded row: bank = (4c + l) % 64
            ldsQ[c * LDSTRIDE + l] = expf(srow[c] - m) * inv;
    }
    __syncthreads();

    // ---- WMMA fp32 GEMM: each wave -> 16 rows x 64 cols ----
    const int lane = threadIdx.x & 31;
    const int wave = threadIdx.x >> 5;
    const int half = lane >> 4;           // 0 or 1
    const int m16  = lane & 15;           // row-in-tile (A) / col-in-tile (B,D)
    const int kofs = half << 1;           // K sub-offset 0 or 2 (32-bit 16x4 layout)

    const int row_base = blockIdx.x * ROWS_PER_BLOCK + wave * 16;
    const float* aptr = mu + (row_base + m16) * L_DIM + kofs;

    const float* b0 = &ldsQ[( 0 + m16) * LDSTRIDE + kofs];
    const float* b1 = &ldsQ[(16 + m16) * LDSTRIDE + kofs];
    const float* b2 = &ldsQ[(32 + m16) * LDSTRIDE + kofs];
    const float* b3 = &ldsQ[(48 + m16) * LDSTRIDE + kofs];

    v8f acc0 = {}, acc1 = {}, acc2 = {}, acc3 = {};

    #pragma unroll 4
    for (int k = 0; k < L_DIM; k += 4) {
        v2f a = *(const v2f*)(aptr + k);          // global_load_b64, A frag (16x4 f32)
        a.x = fmaxf(a.x, 1e-12f);                 // clip(mu, EPS)
        a.y = fmaxf(a.y, 1e-12f);
        v2f f0 = *(const v2f*)(b0 + k);           // ds_load_b64, B frags (4x16 f32)
        v2f f1 = *(const v2f*)(b1 + k);
        v2f f2 = *(const v2f*)(b2 + k);
        v2f f3 = *(const v2f*)(b3 + k);
        acc0 = __builtin_amdgcn_wmma_f32_16x16x4_f32(false, a, false, f0, (short)0, acc0, false, false);
        acc1 = __builtin_amdgcn_wmma_f32_16x16x4_f32(false, a, false, f1, (short)0, acc1, false, false);
        acc2 = __builtin_amdgcn_wmma_f32_16x16x4_f32(false, a, false, f2, (short)0, acc2, false, false);
        acc3 = __builtin_amdgcn_wmma_f32_16x16x4_f32(false, a, false, f3, (short)0, acc3, false, false);
    }

    // ---- Epilogue: y = log(acc); D layout: VGPR r -> row r + 8*half, col = m16 ----
    float* op = out + (row_base + (half << 3)) * C_DIM + m16;
    #pragma unroll
    for (int r = 0; r < 8; ++r) {
        op[r * C_DIM +  0] = logf(acc0[r]);
        op[r * C_DIM + 16] = logf(acc1[r]);
        op[r * C_DIM + 32] = logf(acc2[r]);
        op[r * C_DIM + 48] = logf(acc3[r]);
    }
}

extern "C" void kernel_launch(void* const* d_in, const int* in_sizes, int n_in,
                              void* d_out, int out_size, void* d_ws, size_t ws_size,
                              hipStream_t stream) {
    const float* mu   = (const float*)d_in[0];   // (32768, 256)
    const float* leaf = (const float*)d_in[1];   // (256, 64)
    float* out = (float*)d_out;                  // (32768, 64)
    (void)in_sizes; (void)n_in; (void)out_size; (void)d_ws; (void)ws_size;

    const size_t lds_bytes = (size_t)C_DIM * LDSTRIDE * sizeof(float);  // 66560 B < 320 KB/WGP
    LeafLogMixtureHead_kernel<<<N_ROWS / ROWS_PER_BLOCK, 256, lds_bytes, stream>>>(mu, leaf, out);
}